// AnchorLoss_10471130267820
// MI455X (gfx1250) — compile-verified
//
#include <hip/hip_runtime.h>

typedef __attribute__((ext_vector_type(2))) float v2f;
typedef __attribute__((ext_vector_type(8))) float v8f;

#define B_ 8
#define N_ 2048
#define BN (B_ * N_)                 // 16384 points
#define TILE 16
#define NT (N_ / TILE)               // 128 tiles per row
#define TILES_PER_B (NT * NT)        // 16384 tiles per batch
#define WAVES_PER_BLOCK 8
#define MAIN_BLOCKS (B_ * TILES_PER_B / WAVES_PER_BLOCK)  // 16384 blocks
#define INV_TEMP 0.1f

// Stage 1: pos = embedding + abs_coords ; norm = |pos|^2
__global__ void anchor_prep(const float* __restrict__ emb,
                            const float* __restrict__ coords,
                            float2* __restrict__ pos,
                            float* __restrict__ norm) {
  int idx = blockIdx.x * blockDim.x + threadIdx.x;
  if (idx < BN) {
    float x = emb[2 * idx + 0] + coords[2 * idx + 0];
    float y = emb[2 * idx + 1] + coords[2 * idx + 1];
    pos[idx] = make_float2(x, y);
    norm[idx] = x * x + y * y;
  }
}

// Stage 2: one wave computes one 16x16 tile of the gram matrix via
// V_WMMA_F32_16X16X4_F32 (K=2 padded to 4 with zeros), then the masked
// nonlinear-distance epilogue. Per-block partial sums (deterministic order).
__global__ void anchor_tiles(const float2* __restrict__ pos,
                             const float* __restrict__ norm,
                             const int* __restrict__ mask,
                             float* __restrict__ partials) {
  const int lane = threadIdx.x & 31;
  const int wave = threadIdx.x >> 5;
  const int gw   = blockIdx.x * WAVES_PER_BLOCK + wave;   // global tile id
  const int b    = gw / TILES_PER_B;
  const int r    = gw % TILES_PER_B;
  const int i0   = (r / NT) * TILE;
  const int j0   = (r % NT) * TILE;
  const int bN   = b * N_;

  // A (16x4) and B (4x16) layouts coincide for K<=2: lanes 0-15 carry {x,y}
  // of the 16 rows; lanes 16-31 carry the K=2,3 zero padding.
  v2f a = {0.f, 0.f};
  v2f bm = {0.f, 0.f};
  if (lane < 16) {
    float2 pa = pos[bN + i0 + lane];
    float2 pb = pos[bN + j0 + lane];
    a[0] = pa.x;  a[1] = pa.y;
    bm[0] = pb.x; bm[1] = pb.y;
  }
  v8f cz = {};
  // gram[v][lane]: row M = v + 8*(lane>=16), col N = lane & 15
  v8f g = __builtin_amdgcn_wmma_f32_16x16x4_f32(
      /*neg_a=*/false, a, /*neg_b=*/false, bm,
      /*c_mod=*/(short)0, cz, /*reuse_a=*/false, /*reuse_b=*/false);

  const int   hi   = lane >> 4;
  const int   col  = lane & 15;
  const float nj   = norm[bN + j0 + col];
  const long long row0 = (long long)(bN + i0 + 8 * hi) * N_ + (j0 + col);

  float acc = 0.f;
#pragma unroll
  for (int v = 0; v < 8; ++v) {
    const int   i  = i0 + v + 8 * hi;
    const float ni = norm[bN + i];
    const float sq = fmaf(-2.f, g[v], ni + nj);      // |pi|^2+|pj|^2-2 pi.pj
    const float val = 1.f - __expf(-sq * INV_TEMP);
    const int   m   = mask[row0 + (long long)v * N_];
    acc += (m == 1) ? val : 0.f;
  }

  // wave32 butterfly reduction
#pragma unroll
  for (int off = 16; off > 0; off >>= 1)
    acc += __shfl_xor(acc, off, 32);

  __shared__ float wsum[WAVES_PER_BLOCK];
  if (lane == 0) wsum[wave] = acc;
  __syncthreads();
  if (threadIdx.x == 0) {
    float s = 0.f;
#pragma unroll
    for (int w = 0; w < WAVES_PER_BLOCK; ++w) s += wsum[w];
    partials[blockIdx.x] = s;
  }
}

// Stage 3: deterministic single-block reduction of 16384 partials.
__global__ void anchor_reduce(const float* __restrict__ partials,
                              float* __restrict__ out) {
  __shared__ float sm[256];
  float s = 0.f;
  for (int t = threadIdx.x; t < MAIN_BLOCKS; t += 256) s += partials[t];
  sm[threadIdx.x] = s;
  __syncthreads();
  for (int off = 128; off > 0; off >>= 1) {
    if ((int)threadIdx.x < off) sm[threadIdx.x] += sm[threadIdx.x + off];
    __syncthreads();
  }
  if (threadIdx.x == 0) out[0] = sm[0];
}

extern "C" void kernel_launch(void* const* d_in, const int* in_sizes, int n_in,
                              void* d_out, int out_size, void* d_ws, size_t ws_size,
                              hipStream_t stream) {
  const float* emb    = (const float*)d_in[0];   // [B,N,2] f32
  const float* coords = (const float*)d_in[1];   // [B,N,2] f32
  const int*   mask   = (const int*)d_in[2];     // [B,N,N] i32
  float*       out    = (float*)d_out;           // scalar f32

  // Workspace layout: float2 pos[BN] | float norm[BN] | float partials[MAIN_BLOCKS]
  float2* pos      = (float2*)d_ws;
  float*  norm     = (float*)d_ws + 2 * BN;
  float*  partials = norm + BN;

  anchor_prep<<<BN / 256, 256, 0, stream>>>(emb, coords, pos, norm);
  anchor_tiles<<<MAIN_BLOCKS, 256, 0, stream>>>(pos, norm, mask, partials);
  anchor_reduce<<<1, 256, 0, stream>>>(partials, out);
}